// GNNLayer_60335700574605
// MI455X (gfx1250) — compile-verified
//
#include <hip/hip_runtime.h>
#include <hip/hip_bf16.h>

// GNN layer for MI455X (gfx1250, wave32):
//   new_H = ReLU(H @ W^T)                  [N,128]  -> d_ws rows 0..N-1
//   d_ws row N = column-min of new_H       (sentinel row, via u32 atomic min)
//   out[n] = max over K neighbour rows of padded (index N hits sentinel)

#define DI 128
#define DO 128
#define LDS_STRIDE 132   // 16-row H tile, padded so ds_load_b64 is bank-conflict free

typedef __attribute__((ext_vector_type(2))) float v2f;
typedef __attribute__((ext_vector_type(8))) float v8f;

// ---------------------------------------------------------------------------
// Kernel 0: initialize sentinel row (row N of padded) to +inf so the u32
// atomicMin of non-negative (post-ReLU) floats computes the true column min.
// ---------------------------------------------------------------------------
__global__ void gnn_init_minrow(float* __restrict__ padded, int N) {
    int o = threadIdx.x;            // launched with 128 threads
    padded[(size_t)N * DO + o] = __uint_as_float(0x7F800000u);  // +inf
}

// ---------------------------------------------------------------------------
// Kernel 1: WMMA f32 16x16x4 GEMM + ReLU + column-min.
// Block = 256 threads = 8 waves. Block owns H rows [row0, row0+16).
// Wave w owns output columns [16w, 16w+16).  K loop: 128/4 = 32 WMMA steps.
//
// f32 WMMA operand layout (ISA 7.12.2):
//  A (16x4): lanes 0-15 = rows 0-15; v0={K=k0 | K=k0+2}, v1={K=k0+1 | K=k0+3}
//  B (4x16): lanes hold N=0-15; same K split across lane halves (B[k][n]=W[n][k])
//  D (16x16): VGPR j: lanes 0-15 -> M=j, lanes 16-31 -> M=j+8; N = lane&15
// ---------------------------------------------------------------------------
__global__ void __launch_bounds__(256)
gnn_gemm_relu_min(const float* __restrict__ H,
                  const float* __restrict__ W,
                  float* __restrict__ padded,   // (N+1) x 128
                  int N) {
    __shared__ float Hs[16 * LDS_STRIDE];

    const int tid  = threadIdx.x;
    const int wave = tid >> 5;          // 0..7
    const int lane = tid & 31;
    const int half = lane >> 4;         // 0: K even pair / rows j; 1: K+2 / rows j+8
    const int l16  = lane & 15;
    const int row0 = blockIdx.x * 16;
    const int nb   = wave * 16;         // output column base for this wave

    // Stage H tile [16 x 128] -> LDS (2048 floats; 8 per thread, float4 x2).
    {
        const int r = tid >> 4;             // 0..15
        const int c = (tid & 15) * 8;       // 0,8,...,120
        int gr = row0 + r;
        if (gr >= N) gr = N - 1;            // clamp (N is a multiple of 16 here)
        const float4* src = (const float4*)(H + (size_t)gr * DI + c);
        float4 a0 = src[0];
        float4 a1 = src[1];
        *(float4*)&Hs[r * LDS_STRIDE + c]     = a0;   // 528B row pitch: 16B aligned
        *(float4*)&Hs[r * LDS_STRIDE + c + 4] = a1;
    }
    __syncthreads();

    // Per-lane fragment base pointers.
    const float* hs_base = &Hs[l16 * LDS_STRIDE + 2 * half];            // A: H row l16
    const float* wb_base = W + (size_t)(nb + l16) * DI + 2 * half;      // B: W row nb+l16

    v8f acc = {0.f, 0.f, 0.f, 0.f, 0.f, 0.f, 0.f, 0.f};

    #pragma unroll
    for (int kk = 0; kk < DI / 4; ++kk) {
        const int k0 = kk * 4;
        v2f a = *(const v2f*)(hs_base + k0);   // ds_load_b64, conflict-free
        v2f b = *(const v2f*)(wb_base + k0);   // global_load_b64, L2/WGP$ resident
        acc = __builtin_amdgcn_wmma_f32_16x16x4_f32(
                  /*neg_a=*/false, a, /*neg_b=*/false, b,
                  /*c_mod=*/(short)0, acc, /*reuse_a=*/false, /*reuse_b=*/false);
    }

    // ReLU + per-lane column min over the 8 accumulator rows.
    float colmin = __uint_as_float(0x7F800000u);
    #pragma unroll
    for (int j = 0; j < 8; ++j) {
        acc[j] = fmaxf(acc[j], 0.0f);
        colmin = fminf(colmin, acc[j]);
    }
    // Combine the two lane-halves that share a column (lanes L and L+16).
    colmin = fminf(colmin, __shfl_xor(colmin, 16));

    // Post-ReLU values are >= 0: u32 ordering == float ordering.
    if (half == 0) {
        atomicMin((unsigned int*)(padded + (size_t)N * DO + nb + l16),
                  __float_as_uint(colmin));
    }

    // Store D fragment: lane half 0 -> rows row0+j, half 1 -> rows row0+8+j.
    #pragma unroll
    for (int j = 0; j < 8; ++j) {
        const int r = row0 + half * 8 + j;
        if (r < N) padded[(size_t)r * DO + nb + l16] = acc[j];
    }
}

// ---------------------------------------------------------------------------
// Kernel 2: neighbour gather + max-pool. One node per wave32; each lane owns
// one float4 of the 128-wide row (512B fully-coalesced row reads, L2-resident
// since padded is 51 MB < 192 MB L2). Prefetch next neighbour row to overlap
// L2 latency (global_prefetch_b8).
// ---------------------------------------------------------------------------
__global__ void __launch_bounds__(256)
gnn_gather_max(const long long* __restrict__ nbr,   // [N, K] int64 in [0, N]
               const float* __restrict__ padded,    // (N+1) x 128
               float* __restrict__ out,             // [N, 128]
               int N, int K) {
    const int wave = threadIdx.x >> 5;
    const int lane = threadIdx.x & 31;
    const int node = blockIdx.x * 8 + wave;
    if (node >= N) return;

    const long long* idx = nbr + (size_t)node * K;

    const float ninf = -__builtin_inff();
    float4 m = make_float4(ninf, ninf, ninf, ninf);

    long long jn = idx[0];
    __builtin_prefetch(padded + (size_t)jn * DO + lane * 4, 0, 3);

    for (int k = 0; k < K; ++k) {
        const long long j = jn;
        if (k + 1 < K) {
            jn = idx[k + 1];
            __builtin_prefetch(padded + (size_t)jn * DO + lane * 4, 0, 3);
        }
        const float4 v = ((const float4*)(padded + (size_t)j * DO))[lane];
        m.x = fmaxf(m.x, v.x);
        m.y = fmaxf(m.y, v.y);
        m.z = fmaxf(m.z, v.z);
        m.w = fmaxf(m.w, v.w);
    }
    ((float4*)(out + (size_t)node * DO))[lane] = m;
}

// ---------------------------------------------------------------------------
extern "C" void kernel_launch(void* const* d_in, const int* in_sizes, int n_in,
                              void* d_out, int out_size, void* d_ws, size_t ws_size,
                              hipStream_t stream) {
    const float*     H   = (const float*)d_in[0];       // [N, 128] fp32
    const long long* nbr = (const long long*)d_in[1];   // [N, K]   int64
    const float*     W   = (const float*)d_in[2];       // [128, 128] fp32
    float*           out = (float*)d_out;               // [N, 128] fp32

    const int N = in_sizes[0] / DI;
    const int K = in_sizes[1] / N;

    float* padded = (float*)d_ws;   // (N+1) x 128 floats

    gnn_init_minrow<<<1, DO, 0, stream>>>(padded, N);

    const int gemm_blocks = (N + 15) / 16;
    gnn_gemm_relu_min<<<gemm_blocks, 256, 0, stream>>>(H, W, padded, N);

    const int gather_blocks = (N + 7) / 8;
    gnn_gather_max<<<gather_blocks, 256, 0, stream>>>(nbr, padded, out, N, K);
}